// EncoderGRU_81028853006912
// MI455X (gfx1250) — compile-verified
//
#include <hip/hip_runtime.h>
#include <hip/hip_bf16.h>

#define BSZ   32
#define TLEN  128
#define HID   1024
#define G3    (3 * HID)
#define VOCAB 32000

typedef float v2f __attribute__((ext_vector_type(2)));
typedef float v8f __attribute__((ext_vector_type(8)));

// ---------------------------------------------------------------------------
// Kernel A: gx[b,t,g] = W_ih[g, x[b,t]] + b_ih[g]
// One block per (b,t) token; coalesced writes, strided gather reads (inherent).
// ---------------------------------------------------------------------------
__global__ void gru_embed(const int* __restrict__ x,
                          const float* __restrict__ W_ih,
                          const float* __restrict__ b_ih,
                          float* __restrict__ gx) {
  const int bt  = blockIdx.x;            // 0 .. B*T-1
  const int tok = x[bt];
  const float* col = W_ih + tok;         // column 'tok', row stride = VOCAB
  float* out = gx + (size_t)bt * G3;
  for (int g = threadIdx.x; g < G3; g += blockDim.x) {
    out[g] = col[(size_t)g * VOCAB] + b_ih[g];
  }
}

__global__ void zero_f32(float* __restrict__ p, int n) {
  int i = blockIdx.x * blockDim.x + threadIdx.x;
  if (i < n) p[i] = 0.0f;
}

__global__ void copy_f32(const float* __restrict__ s, float* __restrict__ d, int n) {
  int i = blockIdx.x * blockDim.x + threadIdx.x;
  if (i < n) d[i] = s[i];
}

// ---------------------------------------------------------------------------
// Kernel B: one GRU time step.
//   gh = h_in @ W_hh^T + b_hh   (fp32 WMMA 16x16x4, K=1024, 4-way K-split)
//   LDS reduction of K partials, then gate math + write h_out and outs[:,t,:]
//
// Grid: 64 blocks (16 cols each) x 256 threads (8 waves).
//   wave = ks*2 + m :  m  = m-tile (batch rows 0-15 / 16-31)
//                      ks = K-slice (256 wide each)
// Each wave runs a 64-chunk serial K-loop (4x shorter than unsplit),
// partials are summed in fixed order via LDS for determinism.
// ---------------------------------------------------------------------------
__global__ __launch_bounds__(256) void gru_step(
    const float* __restrict__ h_in,    // B x H
    const float* __restrict__ W_hh,    // 3H x H, row-major
    const float* __restrict__ b_hh,    // 3H
    const float* __restrict__ gx,      // B x T x 3H (b_ih already added)
    float* __restrict__ h_out,         // B x H
    float* __restrict__ outbuf,        // B x T x H
    int t) {
  __shared__ float red[8][32][25];         // [wave][lane][3*8], pad->no bank conflicts

  const int tid    = threadIdx.x;
  const int waveId = tid >> 5;             // 0..7
  const int m      = waveId & 1;           // m-tile: 0 or 1
  const int ks     = waveId >> 1;          // K-slice: 0..3
  const int lane   = tid & 31;
  const int lo     = lane & 15;
  const int hi     = lane >> 4;            // 0 or 1
  const int c0     = blockIdx.x * 16;      // column base within each gate
  const int khalf  = hi * 2;               // K sub-offset per ISA A/B layout
  const int kbase  = ks * (HID / 4);       // 256-wide K slice

  // A: h_in[row = m*16+lo, kbase+k0+khalf .. +1]  (16x4 fp32 A layout)
  const float* Arow = h_in + (size_t)(m * 16 + lo) * HID + kbase + khalf;
  // B: W_hh^T[k, col] = W_hh[gate*H + col, k]     (4x16 fp32 B layout)
  const float* Br = W_hh + (size_t)(c0 + lo) * HID + kbase + khalf;
  const float* Bz = Br + (size_t)HID * HID;
  const float* Bn = Bz + (size_t)HID * HID;

  v8f accR = {}, accZ = {}, accN = {};

#pragma unroll 4
  for (int k0 = 0; k0 < HID / 4; k0 += 4) {
    v2f a  = *(const v2f*)(Arow + k0);
    v2f br = *(const v2f*)(Br + k0);
    v2f bz = *(const v2f*)(Bz + k0);
    v2f bn = *(const v2f*)(Bn + k0);
    // (neg_a, A, neg_b, B, c_mod, C, reuse_a, reuse_b)
    accR = __builtin_amdgcn_wmma_f32_16x16x4_f32(false, a, false, br,
                                                 (short)0, accR, false, false);
    accZ = __builtin_amdgcn_wmma_f32_16x16x4_f32(false, a, false, bz,
                                                 (short)0, accZ, false, false);
    accN = __builtin_amdgcn_wmma_f32_16x16x4_f32(false, a, false, bn,
                                                 (short)0, accN, false, false);
  }

  // Publish partials to LDS
  {
    float* myred = &red[waveId][lane][0];
#pragma unroll
    for (int p = 0; p < 8; ++p) {
      myred[p]      = accR[p];
      myred[8 + p]  = accZ[p];
      myred[16 + p] = accN[p];
    }
  }
  __syncthreads();
  if (ks != 0) return;                     // waves 0,1 finish the step

  // Fixed-order reduction over K-slices 1..3 (deterministic)
#pragma unroll
  for (int s = 1; s < 4; ++s) {
    const float* o = &red[s * 2 + m][lane][0];
#pragma unroll
    for (int p = 0; p < 8; ++p) {
      accR[p] += o[p];
      accZ[p] += o[8 + p];
      accN[p] += o[16 + p];
    }
  }

  // Gate math. C/D layout: VGPR p -> row (p or p+8 by lane half), lane&15 -> col.
  const int   i   = c0 + lo;               // hidden index
  const float bhr = b_hh[i];
  const float bhz = b_hh[HID + i];
  const float bhn = b_hh[2 * HID + i];

#pragma unroll
  for (int p = 0; p < 8; ++p) {
    const int row = m * 16 + p + hi * 8;   // batch index
    const size_t gxbase = ((size_t)row * TLEN + t) * G3 + i;
    const float xr = gx[gxbase];
    const float xz = gx[gxbase + HID];
    const float xn = gx[gxbase + 2 * HID];
    const float hp = h_in[(size_t)row * HID + i];

    const float hr = accR[p] + bhr;
    const float hz = accZ[p] + bhz;
    const float hn = accN[p] + bhn;

    const float r = 1.0f / (1.0f + __expf(-(xr + hr)));
    const float z = 1.0f / (1.0f + __expf(-(xz + hz)));
    const float n = tanhf(xn + r * hn);
    const float hnew = (1.0f - z) * n + z * hp;

    h_out[(size_t)row * HID + i] = hnew;
    outbuf[((size_t)row * TLEN + t) * HID + i] = hnew;
  }
}

// ---------------------------------------------------------------------------
extern "C" void kernel_launch(void* const* d_in, const int* in_sizes, int n_in,
                              void* d_out, int out_size, void* d_ws, size_t ws_size,
                              hipStream_t stream) {
  const int*   x    = (const int*)d_in[0];
  const float* W_ih = (const float*)d_in[1];
  const float* W_hh = (const float*)d_in[2];
  const float* b_ih = (const float*)d_in[3];
  const float* b_hh = (const float*)d_in[4];

  float* out    = (float*)d_out;                   // (B, T, H)
  float* hidden = out + (size_t)BSZ * TLEN * HID;  // (1, B, H)

  // Workspace: gx (B*T*3H) | hbuf0 (B*H) | hbuf1 (B*H)
  float* gx    = (float*)d_ws;
  float* hbuf0 = gx + (size_t)BSZ * TLEN * G3;
  float* hbuf1 = hbuf0 + (size_t)BSZ * HID;

  // 1) Embedding gather (+ b_ih)
  gru_embed<<<BSZ * TLEN, 256, 0, stream>>>(x, W_ih, b_ih, gx);

  // 2) h0 = 0
  zero_f32<<<(BSZ * HID + 255) / 256, 256, 0, stream>>>(hbuf0, BSZ * HID);

  // 3) 128 serial GRU steps, double-buffered hidden state
  for (int t = 0; t < TLEN; ++t) {
    const float* hin  = (t & 1) ? hbuf1 : hbuf0;
    float*       hout = (t & 1) ? hbuf0 : hbuf1;
    gru_step<<<HID / 16, 256, 0, stream>>>(hin, W_hh, b_hh, gx, hout, out, t);
  }

  // 4) hidden = h after last step (T even -> lives in hbuf0)
  copy_f32<<<(BSZ * HID + 255) / 256, 256, 0, stream>>>(hbuf0, hidden, BSZ * HID);
}